// MeshPool_Parrallel_86474871537964
// MI455X (gfx1250) — compile-verified
//
#include <hip/hip_runtime.h>
#include <hip/hip_bf16.h>

// Problem constants (from reference setup_inputs)
constexpr int B = 8;
constexpr int F = 8192;
constexpr int C = 256;
constexpr int K = 4;
constexpr int P = F / 4;          // faces collapsed per batch

typedef float v2f __attribute__((ext_vector_type(2)));
typedef float v8f __attribute__((ext_vector_type(8)));

// ---------------------------------------------------------------------------
// Kernel 1: per-face saliency w^2 = || mean_k features[ring_k] ||^2
// One wave32 handles 16 faces. The mean vector's C=256 entries are reduced
// with V_WMMA_F32_16X16X4_F32: lane p holds m[p%16][4t + 2*(p>>4) + {0,1}]
// which is simultaneously the A(16x4) and B(4x16) operand layout, so
// D += A x A^T accumulates the Gram matrix; diag(D) = ||m||^2 in full f32.
// ---------------------------------------------------------------------------
__global__ __launch_bounds__(256) void k_weights(const float* __restrict__ features,
                                                 const int*   __restrict__ ring,
                                                 float*       __restrict__ w2out) {
    __shared__ float diag[8 * 16];                 // 8 waves * 16 faces
    const int lane = threadIdx.x & 31;
    const int wv   = threadIdx.x >> 5;
    const int gw   = blockIdx.x * 8 + wv;          // global wave id
    const int b    = gw >> 9;                      // F/16 = 512 waves per batch
    const int tile = gw & 511;
    const int face = tile * 16 + (lane & 15);
    const int half = lane >> 4;                    // 0: c%4 in {0,1}; 1: {2,3}

    const long base = (long)b * F + face;
    const int4 r = *(const int4*)(ring + base * K);
    const float* f0 = features + ((long)b * F + r.x) * C;
    const float* f1 = features + ((long)b * F + r.y) * C;
    const float* f2 = features + ((long)b * F + r.z) * C;
    const float* f3 = features + ((long)b * F + r.w) * C;

    v8f acc = {};
    #pragma unroll 4
    for (int t = 0; t < C / 4; ++t) {
        const int c = 4 * t + 2 * half;
        const float2 a0 = *(const float2*)(f0 + c);
        const float2 a1 = *(const float2*)(f1 + c);
        const float2 a2 = *(const float2*)(f2 + c);
        const float2 a3 = *(const float2*)(f3 + c);
        v2f m;
        m.x = 0.25f * ((a0.x + a1.x) + (a2.x + a3.x));
        m.y = 0.25f * ((a0.y + a1.y) + (a2.y + a3.y));
        // D += A * B + C   (A == B per-lane registers -> Gram matrix)
        acc = __builtin_amdgcn_wmma_f32_16x16x4_f32(false, m, false, m,
                                                    (short)0, acc, false, false);
    }

    // diag(D): element (i,i) sits at lane i, VGPR i (i<8) or lane 16+i, VGPR i-8
    #pragma unroll
    for (int v = 0; v < 8; ++v) {
        if (lane == v)      diag[wv * 16 + v]     = acc[v];
        if (lane == v + 24) diag[wv * 16 + v + 8] = acc[v];
    }
    __syncthreads();
    if (lane < 16) w2out[base] = diag[wv * 16 + lane];
}

// ---------------------------------------------------------------------------
// Kernel 2: per-batch selection of the P smallest w^2 (lowest index on ties,
// matching top_k). One 1024-thread workgroup per batch; all 8192 values in
// LDS. w^2 >= 0 so uint bit order == float order. Binary search the P-th
// smallest, then an ordered prefix-scan resolves ties in face-index order.
// ---------------------------------------------------------------------------
__global__ __launch_bounds__(1024) void k_select(const float* __restrict__ w2,
                                                 int*         __restrict__ mask) {
    __shared__ unsigned int vals[F];     // 32 KB
    __shared__ int scan[1024];
    __shared__ int cnt;
    const int b   = blockIdx.x;
    const int tid = threadIdx.x;
    const unsigned int* wbits = (const unsigned int*)(w2 + (long)b * F);

    for (int i = tid; i < F; i += 1024) vals[i] = wbits[i];
    __syncthreads();

    // minimal T with countLE(T) >= P
    unsigned int lo = 0u, hi = 0xFFFFFFFFu;
    for (int it = 0; it < 32; ++it) {
        const unsigned int mid = lo + ((hi - lo) >> 1);
        if (tid == 0) cnt = 0;
        __syncthreads();
        int local = 0;
        #pragma unroll
        for (int j = 0; j < 8; ++j) local += (vals[tid * 8 + j] <= mid) ? 1 : 0;
        atomicAdd(&cnt, local);
        __syncthreads();
        const int c = cnt;
        __syncthreads();
        if (c >= P) hi = mid; else lo = mid + 1;   // uniform across block
    }
    const unsigned int T = lo;

    // count strictly-less -> tie quota
    if (tid == 0) cnt = 0;
    __syncthreads();
    {
        int local = 0;
        #pragma unroll
        for (int j = 0; j < 8; ++j) local += (vals[tid * 8 + j] < T) ? 1 : 0;
        atomicAdd(&cnt, local);
    }
    __syncthreads();
    const int quota = P - cnt;
    __syncthreads();

    // ordered tie selection: inclusive scan of per-thread tie counts
    int tie = 0;
    #pragma unroll
    for (int j = 0; j < 8; ++j) tie += (vals[tid * 8 + j] == T) ? 1 : 0;
    scan[tid] = tie;
    __syncthreads();
    for (int off = 1; off < 1024; off <<= 1) {
        const int add = (tid >= off) ? scan[tid - off] : 0;
        __syncthreads();
        scan[tid] += add;
        __syncthreads();
    }
    int run = scan[tid] - tie;                     // exclusive prefix of ties
    int* mb = mask + (long)b * F;
    #pragma unroll
    for (int j = 0; j < 8; ++j) {
        const int f = tid * 8 + j;
        const unsigned int u = vals[f];
        int sel = 0;
        if (u < T) sel = 1;
        else if (u == T && run < quota) { sel = 1; ++run; }
        mb[f] = sel;
    }
}

// ---------------------------------------------------------------------------
// Kernel 3: scatter cf/3 into the 3 adjacent faces, + neighbor counts.
// One block per face (C channels); blocks for non-collapsed faces exit.
// ---------------------------------------------------------------------------
__global__ __launch_bounds__(256) void k_scatter(const float* __restrict__ features,
                                                 const int*   __restrict__ adj,
                                                 const int*   __restrict__ mask,
                                                 float*       __restrict__ scat,   // = d_out (zeroed)
                                                 float*       __restrict__ cnt) {
    const int gface = blockIdx.x;                  // b*F + f
    if (!mask[gface]) return;
    const int  b  = gface >> 13;                   // / F
    const int  c  = threadIdx.x;
    const long nb = (long)b * F;
    const int n0 = adj[gface * 3 + 0];
    const int n1 = adj[gface * 3 + 1];
    const int n2 = adj[gface * 3 + 2];
    const float val = features[(long)gface * C + c] * (1.0f / 3.0f);
    unsafeAtomicAdd(scat + (nb + n0) * C + c, val);
    unsafeAtomicAdd(scat + (nb + n1) * C + c, val);
    unsafeAtomicAdd(scat + (nb + n2) * C + c, val);
    if (c == 0) {
        unsafeAtomicAdd(cnt + nb + n0, 1.0f);
        unsafeAtomicAdd(cnt + nb + n1, 1.0f);
        unsafeAtomicAdd(cnt + nb + n2, 1.0f);
    }
}

// ---------------------------------------------------------------------------
// Kernel 4: merged = (features + scat) / (1 + cnt), zero collapsed rows.
// float4-vectorized; scat lives in d_out and is overwritten in place.
// ---------------------------------------------------------------------------
__global__ __launch_bounds__(256) void k_final(const float* __restrict__ features,
                                               const float* __restrict__ cnt,
                                               const int*   __restrict__ mask,
                                               float*       __restrict__ out) {
    const long gid = (long)blockIdx.x * 256 + threadIdx.x;  // one float4 each
    const long idx = gid * 4;
    const int  gface = (int)(idx >> 8);                     // / C
    const float keep = mask[gface] ? 0.0f : 1.0f;
    const float den  = 1.0f + cnt[gface];
    const float4 f = *(const float4*)(features + idx);
    const float4 s = *(const float4*)(out + idx);
    float4 r;
    r.x = keep * ((f.x + s.x) / den);
    r.y = keep * ((f.y + s.y) / den);
    r.z = keep * ((f.z + s.z) / den);
    r.w = keep * ((f.w + s.w) / den);
    *(float4*)(out + idx) = r;
}

// ---------------------------------------------------------------------------
extern "C" void kernel_launch(void* const* d_in, const int* in_sizes, int n_in,
                              void* d_out, int out_size, void* d_ws, size_t ws_size,
                              hipStream_t stream) {
    const float* features  = (const float*)d_in[0];   // [B,F,C] f32
    const int*   adjacency = (const int*)d_in[1];     // [B,F,3] i32
    const int*   ring      = (const int*)d_in[2];     // [B,F,K] i32
    float*       out       = (float*)d_out;           // [B,F,C] f32

    char*  ws   = (char*)d_ws;
    float* w2   = (float*)(ws);                        // B*F floats
    int*   mask = (int*)  (ws + (size_t)B * F * 4);    // B*F ints
    float* cntb = (float*)(ws + (size_t)B * F * 8);    // B*F floats

    // d_out doubles as the scatter accumulator; zero it (and counts) each call
    hipMemsetAsync(out,  0, (size_t)B * F * C * sizeof(float), stream);
    hipMemsetAsync(cntb, 0, (size_t)B * F * sizeof(float), stream);

    k_weights<<<(B * F) / 128, 256, 0, stream>>>(features, ring, w2);
    k_select <<<B, 1024, 0, stream>>>(w2, mask);
    k_scatter<<<B * F, C, 0, stream>>>(features, adjacency, mask, out, cntb);
    k_final  <<<(B * F * C / 4) / 256, 256, 0, stream>>>(features, cntb, mask, out);
}